// PointRendSemSegHead_62629213110772
// MI455X (gfx1250) — compile-verified
//
#include <hip/hip_runtime.h>

typedef float v2f __attribute__((ext_vector_type(2)));
typedef float v8f __attribute__((ext_vector_type(8)));

#define NUM_CLASSES 19
#define FC_DIM 256
#define IN_CH 256
#define KPTS 8192
#define KPAD 276            // 275 padded to multiple of 4 for WMMA K-steps
#define MPAD_PRED 128       // pred rows padded to a full block tile -> no A-load guard
#define FEAT_H 256
#define FEAT_W 512
#define COARSE_H 128
#define COARSE_W 256

// ---------------------------------------------------------------------------
// 2x bilinear upsample (half-pixel centers, edge clamp) — memory-bound stream
// ---------------------------------------------------------------------------
__global__ void upsample2x_kernel(const float* __restrict__ in, float* __restrict__ out,
                                  int NC, int H, int W) {
    int OW = 2 * W, OH = 2 * H;
    long long total = (long long)NC * OH * OW;
    long long stride = (long long)gridDim.x * blockDim.x;
    for (long long e = (long long)blockIdx.x * blockDim.x + threadIdx.x; e < total; e += stride) {
        long long hw = (long long)OH * OW;
        int nc = (int)(e / hw);
        long long r = e % hw;
        int y = (int)(r / OW), x = (int)(r % OW);
        float sx = (x + 0.5f) * 0.5f - 0.5f;
        float sy = (y + 0.5f) * 0.5f - 0.5f;
        float fx0 = floorf(sx), fy0 = floorf(sy);
        int x0 = (int)fx0, y0 = (int)fy0;
        float wx = sx - fx0, wy = sy - fy0;
        int x0c = min(max(x0, 0), W - 1), x1c = min(max(x0 + 1, 0), W - 1);
        int y0c = min(max(y0, 0), H - 1), y1c = min(max(y0 + 1, 0), H - 1);
        const float* p = in + (long long)nc * H * W;
        float v = p[(long long)y0c * W + x0c] * (1.f - wx) * (1.f - wy)
                + p[(long long)y0c * W + x1c] * wx * (1.f - wy)
                + p[(long long)y1c * W + x0c] * (1.f - wx) * wy
                + p[(long long)y1c * W + x1c] * wx * wy;
        out[e] = v;
    }
}

// ---------------------------------------------------------------------------
// uncertainty = second_max - max over 19 channels, emit order-preserving key
// ---------------------------------------------------------------------------
__global__ void uncertainty_kernel(const float* __restrict__ sem, unsigned* __restrict__ keys,
                                   long long HW) {
    long long total = 2LL * HW;
    long long stride = (long long)gridDim.x * blockDim.x;
    for (long long e = (long long)blockIdx.x * blockDim.x + threadIdx.x; e < total; e += stride) {
        int b = (int)(e / HW);
        long long i = e % HW;
        const float* p = sem + (long long)b * NUM_CLASSES * HW + i;
        float m1 = -3.4e38f, m2 = -3.4e38f;
        for (int c = 0; c < NUM_CLASSES; ++c) {
            float v = p[(long long)c * HW];
            if (v > m1) { m2 = m1; m1 = v; } else if (v > m2) { m2 = v; }
        }
        unsigned u = __float_as_uint(m2 - m1);
        u = (u & 0x80000000u) ? ~u : (u | 0x80000000u);   // larger float -> larger uint
        keys[e] = u;
    }
}

// ---------------------------------------------------------------------------
// radix-select (4 x 8-bit MSB-first) for exact k-th largest key per batch
// ---------------------------------------------------------------------------
__global__ void reset_select_kernel(unsigned* hist, unsigned* prefix, int* kRem,
                                    int* cntG, int* cntT) {
    int t = threadIdx.x;
    if (t < 512) hist[t] = 0u;
    if (t < 2) { prefix[t] = 0u; kRem[t] = KPTS; cntG[t] = 0; cntT[t] = 0; }
}

__global__ void hist_kernel(const unsigned* __restrict__ keys, long long HW,
                            const unsigned* __restrict__ prefix, unsigned* __restrict__ hist,
                            int pass) {
    __shared__ unsigned sh[512];
    for (int t = threadIdx.x; t < 512; t += blockDim.x) sh[t] = 0u;
    __syncthreads();
    int shift = 24 - 8 * pass;
    long long total = 2LL * HW;
    long long stride = (long long)gridDim.x * blockDim.x;
    for (long long e = (long long)blockIdx.x * blockDim.x + threadIdx.x; e < total; e += stride) {
        int b = (int)(e / HW);
        unsigned key = keys[e];
        bool cand = (pass == 0) || (((key ^ prefix[b]) >> (shift + 8)) == 0u);
        if (cand) atomicAdd(&sh[b * 256 + ((key >> shift) & 255u)], 1u);
    }
    __syncthreads();
    for (int t = threadIdx.x; t < 512; t += blockDim.x)
        if (sh[t]) atomicAdd(&hist[t], sh[t]);
}

__global__ void select_kernel(unsigned* hist, unsigned* prefix, int* kRem,
                              unsigned* thresh, int* ties, int pass) {
    int b = threadIdx.x;
    if (b >= 2) return;
    int shift = 24 - 8 * pass;
    int k = kRem[b];
    unsigned cum = 0; int chosen = 0;
    for (int d = 255; d >= 0; --d) {
        unsigned c = hist[b * 256 + d];
        if (cum + c >= (unsigned)k) { chosen = d; break; }
        cum += c;
    }
    prefix[b] |= ((unsigned)chosen) << shift;
    kRem[b] = k - (int)cum;
    if (pass == 3) { thresh[b] = prefix[b]; ties[b] = k - (int)cum; }
    for (int d = 0; d < 256; ++d) hist[b * 256 + d] = 0u;  // ready for next pass
}

__global__ void collect_kernel(const unsigned* __restrict__ keys, long long HW,
                               const unsigned* __restrict__ thresh, const int* __restrict__ ties,
                               int* __restrict__ cntG, int* __restrict__ cntT,
                               int* __restrict__ idx) {
    long long total = 2LL * HW;
    long long stride = (long long)gridDim.x * blockDim.x;
    for (long long e = (long long)blockIdx.x * blockDim.x + threadIdx.x; e < total; e += stride) {
        int b = (int)(e / HW);
        long long i = e % HW;
        unsigned key = keys[e];
        unsigned T = thresh[b];
        if (key > T) {
            int pos = atomicAdd(&cntG[b], 1);
            idx[b * KPTS + pos] = (int)i;
        } else if (key == T) {
            int t = atomicAdd(&cntT[b], 1);
            int nt = ties[b];
            if (t < nt) idx[b * KPTS + (KPTS - nt) + t] = (int)i;
        }
    }
}

// ---------------------------------------------------------------------------
// bilinear point sampling with zero-padding out of bounds (matches reference)
// ---------------------------------------------------------------------------
__device__ __forceinline__ float fetchZ(const float* b, int W, int H, int x, int y) {
    if (x < 0 || x >= W || y < 0 || y >= H) return 0.f;
    return b[(long long)y * W + x];
}
__device__ __forceinline__ float bilinZ(const float* base, int W, int H, float cx, float cy) {
    float x = cx * W - 0.5f, y = cy * H - 0.5f;
    float fx0 = floorf(x), fy0 = floorf(y);
    int x0 = (int)fx0, y0 = (int)fy0;
    float wx = x - fx0, wy = y - fy0;
    float v00 = fetchZ(base, W, H, x0, y0);
    float v01 = fetchZ(base, W, H, x0 + 1, y0);
    float v10 = fetchZ(base, W, H, x0, y0 + 1);
    float v11 = fetchZ(base, W, H, x0 + 1, y0 + 1);
    return v00 * (1.f - wx) * (1.f - wy) + v01 * wx * (1.f - wy)
         + v10 * (1.f - wx) * wy + v11 * wx * wy;
}

// blockIdx.x = point, blockIdx.y = batch, thread = channel
__global__ void __launch_bounds__(256) point_sample_kernel(
        const int* __restrict__ idx, const float* __restrict__ feats,
        const float* __restrict__ coarse, float* __restrict__ XA, float* __restrict__ XB,
        int Ws, int Hs) {
    int p = blockIdx.x, b = blockIdx.y, c = threadIdx.x;
    int i = idx[b * KPTS + p];
    int xs = i % Ws, ys = i / Ws;
    float cx = (xs + 0.5f) / (float)Ws;
    float cy = (ys + 0.5f) / (float)Hs;
    float v = bilinZ(feats + ((long long)b * IN_CH + c) * (FEAT_H * FEAT_W), FEAT_W, FEAT_H, cx, cy);
    XA[((long long)b * KPAD + c) * KPTS + p] = v;
    if (c < NUM_CLASSES) {
        float cv = bilinZ(coarse + ((long long)b * NUM_CLASSES + c) * (COARSE_H * COARSE_W),
                          COARSE_W, COARSE_H, cx, cy);
        long long co = ((long long)b * KPAD + IN_CH + c) * KPTS + p;
        XA[co] = cv; XB[co] = cv;
    }
    if (c == 255) {   // padded K row = 0 in both ping-pong buffers
        long long zo = ((long long)b * KPAD + 275) * KPTS + p;
        XA[zo] = 0.f; XB[zo] = 0.f;
    }
}

// ---------------------------------------------------------------------------
// pad weights (K 275->276, pred M 19->128) and biases once
// ---------------------------------------------------------------------------
__global__ void pack_weights_kernel(const float* w1, const float* b1, const float* w2,
                                    const float* b2, const float* w3, const float* b3,
                                    const float* wp, const float* bp,
                                    float* Wpad, float* Bpad) {
    long long fcsz = (long long)FC_DIM * KPAD;
    long long total = 3 * fcsz + (long long)MPAD_PRED * KPAD;
    long long stride = (long long)gridDim.x * blockDim.x;
    for (long long e = (long long)blockIdx.x * blockDim.x + threadIdx.x; e < total; e += stride) {
        if (e < 3 * fcsz) {
            int l = (int)(e / fcsz);
            long long r = e % fcsz;
            int m = (int)(r / KPAD), k = (int)(r % KPAD);
            const float* src = (l == 0) ? w1 : (l == 1) ? w2 : w3;
            Wpad[e] = (k < 275) ? src[m * 275 + k] : 0.f;
        } else {
            long long r = e - 3 * fcsz;
            int m = (int)(r / KPAD), k = (int)(r % KPAD);
            Wpad[e] = (m < NUM_CLASSES && k < 275) ? wp[m * 275 + k] : 0.f;
        }
    }
    for (long long e = (long long)blockIdx.x * blockDim.x + threadIdx.x; e < 768 + MPAD_PRED; e += stride) {
        if (e < 256)      Bpad[e] = b1[e];
        else if (e < 512) Bpad[e] = b2[e - 256];
        else if (e < 768) Bpad[e] = b3[e - 512];
        else { int m = (int)(e - 768); Bpad[e] = (m < NUM_CLASSES) ? bp[m] : 0.f; }
    }
}

// ---------------------------------------------------------------------------
// f32 WMMA GEMM: Out[m][p] = act(sum_k Wp[m][k] * X[k][p] + bias[m])
// block = 8 waves, block tile 128(M) x 64(N); wave tile 32x32 = 4 WMMA accs.
// Wp is padded so every A load is in-bounds -> no divergent guards in the loop.
// ---------------------------------------------------------------------------
__global__ void __launch_bounds__(256) gemm_wmma_kernel(
        const float* __restrict__ Wp, const float* __restrict__ bias,
        const float* __restrict__ X, float* __restrict__ Out,
        int Mout, int doRelu,
        long long xBatchStride, long long outBatchStride) {
    const int P = KPTS;
    int b = blockIdx.z;
    const float* Xb = X + (long long)b * xBatchStride;
    float* Ob = Out + (long long)b * outBatchStride;
    int lane = threadIdx.x & 31;
    int wave = threadIdx.x >> 5;
    int wm = wave & 3, wn = wave >> 2;
    int mBase = blockIdx.x * 128 + wm * 32;
    int nBase = blockIdx.y * 64 + wn * 32;
    int lr = lane & 15;
    bool hi = lane >= 16;

    v8f acc[2][2];
    v8f zero = {0.f, 0.f, 0.f, 0.f, 0.f, 0.f, 0.f, 0.f};
    acc[0][0] = zero; acc[0][1] = zero; acc[1][0] = zero; acc[1][1] = zero;

    // per-lane base pointers; A rows always valid (weights padded to tile size)
    const float* wp0 = Wp + (long long)(mBase + lr) * KPAD + (hi ? 2 : 0);
    const float* wp1 = wp0 + 16LL * KPAD;
    const float* xp0 = Xb + (long long)(hi ? 2 : 0) * P + (nBase + lr);
    const float* xp1 = xp0 + 16;

    for (int k = 0; k < KPAD; k += 4) {
        v2f a0, a1, b0, b1;
        a0.x = wp0[k];     a0.y = wp0[k + 1];
        a1.x = wp1[k];     a1.y = wp1[k + 1];
        b0.x = xp0[(long long)k * P];       b0.y = xp0[(long long)(k + 1) * P];
        b1.x = xp1[(long long)k * P];       b1.y = xp1[(long long)(k + 1) * P];
        acc[0][0] = __builtin_amdgcn_wmma_f32_16x16x4_f32(false, a0, false, b0, (short)0, acc[0][0], false, false);
        acc[0][1] = __builtin_amdgcn_wmma_f32_16x16x4_f32(false, a0, false, b1, (short)0, acc[0][1], false, false);
        acc[1][0] = __builtin_amdgcn_wmma_f32_16x16x4_f32(false, a1, false, b0, (short)0, acc[1][0], false, false);
        acc[1][1] = __builtin_amdgcn_wmma_f32_16x16x4_f32(false, a1, false, b1, (short)0, acc[1][1], false, false);
    }
#pragma unroll
    for (int s = 0; s < 2; ++s) {
#pragma unroll
        for (int t = 0; t < 2; ++t) {
            int col = nBase + t * 16 + lr;
#pragma unroll
            for (int j = 0; j < 8; ++j) {
                int row = mBase + s * 16 + j + (hi ? 8 : 0);
                if (row < Mout) {
                    float v = acc[s][t][j] + bias[row];
                    if (doRelu) v = fmaxf(v, 0.f);
                    Ob[(long long)row * P + col] = v;
                }
            }
        }
    }
}

// ---------------------------------------------------------------------------
// scatter point logits back into sem at selected flat indices
// ---------------------------------------------------------------------------
__global__ void scatter_kernel(const float* __restrict__ logits, const int* __restrict__ idx,
                               float* __restrict__ sem, long long HW) {
    long long total = 2LL * NUM_CLASSES * KPTS;
    long long stride = (long long)gridDim.x * blockDim.x;
    for (long long e = (long long)blockIdx.x * blockDim.x + threadIdx.x; e < total; e += stride) {
        int n = (int)(e / (NUM_CLASSES * KPTS));
        long long r = e % (NUM_CLASSES * KPTS);
        int c = (int)(r / KPTS), p = (int)(r % KPTS);
        int i = idx[n * KPTS + p];
        sem[((long long)n * NUM_CLASSES + c) * HW + i] =
            logits[((long long)n * NUM_CLASSES + c) * KPTS + p];
    }
}

// ---------------------------------------------------------------------------
extern "C" void kernel_launch(void* const* d_in, const int* in_sizes, int n_in,
                              void* d_out, int out_size, void* d_ws, size_t ws_size,
                              hipStream_t stream) {
    const float* coarse = (const float*)d_in[0];   // (2,19,128,256)
    const float* feats  = (const float*)d_in[1];   // (2,256,256,512)
    const float* fc1_w  = (const float*)d_in[2];
    const float* fc1_b  = (const float*)d_in[3];
    const float* fc2_w  = (const float*)d_in[4];
    const float* fc2_b  = (const float*)d_in[5];
    const float* fc3_w  = (const float*)d_in[6];
    const float* fc3_b  = (const float*)d_in[7];
    const float* pred_w = (const float*)d_in[8];
    const float* pred_b = (const float*)d_in[9];
    float* out = (float*)d_out;                    // (2,19,1024,2048)

    char* ws = (char*)d_ws;
    size_t off = 0;
    auto carve = [&](size_t bytes) -> char* {
        char* p = ws + off;
        off += (bytes + 255) & ~(size_t)255;
        return p;
    };
    float*    semA   = (float*)   carve(2ull * NUM_CLASSES * 256 * 512 * 4);
    float*    semB   = (float*)   carve(2ull * NUM_CLASSES * 512 * 1024 * 4);
    unsigned* keys   = (unsigned*)carve(2ull * 1024 * 2048 * 4);
    int*      idx    = (int*)     carve(2ull * KPTS * 4);
    float*    XA     = (float*)   carve(2ull * KPAD * KPTS * 4);
    float*    XB     = (float*)   carve(2ull * KPAD * KPTS * 4);
    float*    LG     = (float*)   carve(2ull * NUM_CLASSES * KPTS * 4);
    float*    Wpad   = (float*)   carve((3ull * FC_DIM * KPAD + (size_t)MPAD_PRED * KPAD) * 4);
    float*    Bpad   = (float*)   carve((768 + MPAD_PRED) * 4);
    unsigned* hist   = (unsigned*)carve(512 * 4);
    unsigned* prefix = (unsigned*)carve(2 * 4);
    int*      kRem   = (int*)     carve(2 * 4);
    unsigned* thresh = (unsigned*)carve(2 * 4);
    int*      ties   = (int*)     carve(2 * 4);
    int*      cntG   = (int*)     carve(2 * 4);
    int*      cntT   = (int*)     carve(2 * 4);

    pack_weights_kernel<<<512, 256, 0, stream>>>(fc1_w, fc1_b, fc2_w, fc2_b, fc3_w, fc3_b,
                                                 pred_w, pred_b, Wpad, Bpad);

    const long long fcW = (long long)FC_DIM * KPAD;
    struct Step { const float* src; int Hin, Win; float* dst; };
    Step steps[3] = { { coarse, 128, 256, semA },
                      { semA,   256, 512, semB },
                      { semB,   512, 1024, out } };

    for (int s = 0; s < 3; ++s) {
        int Hin = steps[s].Hin, Win = steps[s].Win;
        int H = 2 * Hin, W = 2 * Win;
        long long HW = (long long)H * W;
        const float* src = steps[s].src;
        float* dst = steps[s].dst;

        long long upTotal = 2LL * NUM_CLASSES * HW;
        int upBlocks = (int)((upTotal + 255) / 256);
        if (upBlocks > 65535) upBlocks = 65535;
        upsample2x_kernel<<<upBlocks, 256, 0, stream>>>(src, dst, 2 * NUM_CLASSES, Hin, Win);

        int uBlocks = (int)((2 * HW + 255) / 256);
        if (uBlocks > 32768) uBlocks = 32768;
        uncertainty_kernel<<<uBlocks, 256, 0, stream>>>(dst, keys, HW);

        reset_select_kernel<<<1, 512, 0, stream>>>(hist, prefix, kRem, cntG, cntT);
        for (int pass = 0; pass < 4; ++pass) {
            hist_kernel<<<uBlocks, 256, 0, stream>>>(keys, HW, prefix, hist, pass);
            select_kernel<<<1, 64, 0, stream>>>(hist, prefix, kRem, thresh, ties, pass);
        }
        collect_kernel<<<uBlocks, 256, 0, stream>>>(keys, HW, thresh, ties, cntG, cntT, idx);

        point_sample_kernel<<<dim3(KPTS, 2), 256, 0, stream>>>(idx, feats, coarse, XA, XB, W, H);

        long long xbs = (long long)KPAD * KPTS;
        // fc1: XA -> XB
        gemm_wmma_kernel<<<dim3(2, KPTS / 64, 2), 256, 0, stream>>>(
            Wpad + 0 * fcW, Bpad + 0, XA, XB, FC_DIM, 1, xbs, xbs);
        // fc2: XB -> XA
        gemm_wmma_kernel<<<dim3(2, KPTS / 64, 2), 256, 0, stream>>>(
            Wpad + 1 * fcW, Bpad + 256, XB, XA, FC_DIM, 1, xbs, xbs);
        // fc3: XA -> XB
        gemm_wmma_kernel<<<dim3(2, KPTS / 64, 2), 256, 0, stream>>>(
            Wpad + 2 * fcW, Bpad + 512, XA, XB, FC_DIM, 1, xbs, xbs);
        // pred: XB -> LG (19 real rows, weights zero-padded to 128, no relu)
        gemm_wmma_kernel<<<dim3(1, KPTS / 64, 2), 256, 0, stream>>>(
            Wpad + 3 * fcW, Bpad + 768, XB, LG, NUM_CLASSES, 0,
            xbs, (long long)NUM_CLASSES * KPTS);

        scatter_kernel<<<1216, 256, 0, stream>>>(LG, idx, dst, HW);
    }
}